// MultiBoxLoss_45414984187940
// MI455X (gfx1250) — compile-verified
//
#include <hip/hip_runtime.h>
#include <hip/hip_bf16.h>

#define N_IMG 256
#define P_PRI 8732
#define C_CLS 21
#define M_OBJ 24
#define TPB   256

typedef __attribute__((ext_vector_type(2))) float v2f;
typedef __attribute__((ext_vector_type(8))) float v8f;

// Exact f32 wave32 reduction of two per-lane partials using V_WMMA_F32_16X16X4_F32.
// A is a selector: A[0][0]=A[0][1]=1 -> D[0][n] = B[0][n]+B[1][n] = s1_n + s1_{n+16}
//                  A[1][2]=A[1][3]=1 -> D[1][n] = B[2][n]+B[3][n] = s2_n + s2_{n+16}
// (A layout: lane<16 holds A[M=lane][K=0],A[M][1]; lane>=16 holds A[M=lane-16][K=2],A[M][3];
//  B layout: lane<16 -> B[0][lane],B[2][lane]; lane>=16 -> B[1][lane-16],B[3][lane-16].)
// Finish the 16-wide fold with shfl_xor. Result valid in lane 0. Must be called with
// the full wave active (EXEC all ones).
__device__ inline void wave_fold2_wmma(float s1, float s2, float& o1, float& o2) {
    const unsigned lane = threadIdx.x & 31u;
    v2f a;
    const float sel = (lane == 0u || lane == 17u) ? 1.0f : 0.0f;
    a.x = sel; a.y = sel;
    v2f b; b.x = s1; b.y = s2;
    v8f c = {};
    c = __builtin_amdgcn_wmma_f32_16x16x4_f32(false, a, false, b, (short)0, c, false, false);
    float t1 = c[0];   // D[0][lane] for lanes 0..15
    float t2 = c[1];   // D[1][lane] for lanes 0..15
    t1 += __shfl_xor(t1, 8, 32); t2 += __shfl_xor(t2, 8, 32);
    t1 += __shfl_xor(t1, 4, 32); t2 += __shfl_xor(t2, 4, 32);
    t1 += __shfl_xor(t1, 2, 32); t2 += __shfl_xor(t2, 2, 32);
    t1 += __shfl_xor(t1, 1, 32); t2 += __shfl_xor(t2, 1, 32);
    o1 = t1; o2 = t2;
}

// Kernel A: per-image matching + target encode + CE. One block per image.
__global__ void __launch_bounds__(TPB)
mbl_assign_conf(const float* __restrict__ locs,     // (N,P,4)
                const float* __restrict__ scores,   // (N,P,C)
                const float* __restrict__ boxes,    // (N,M,4) xyxy
                const int*   __restrict__ labels,   // (N,M)
                const float* __restrict__ priors,   // (P,4) cxcywh
                float* __restrict__ conf_neg,       // (N,P) out
                float* __restrict__ loc_sum,        // (N)
                float* __restrict__ conf_pos_sum,   // (N)
                int*   __restrict__ npos_out)       // (N)
{
    const int i   = blockIdx.x;
    const int tid = threadIdx.x;

    __shared__ float4 sbox_xy[M_OBJ];
    __shared__ float4 sbox_c[M_OBJ];
    __shared__ float  sarea[M_OBJ];
    __shared__ int    slab[M_OBJ];
    __shared__ unsigned long long bestkey[M_OBJ];
    __shared__ unsigned char obj_s[P_PRI];
    __shared__ unsigned char cls_s[P_PRI];
    __shared__ float red1[TPB];
    __shared__ float red2[TPB];
    __shared__ int   s_npos;

    if (tid < M_OBJ) {
        const float4 bb = ((const float4*)(boxes + ((size_t)i * M_OBJ + tid) * 4))[0];
        sbox_xy[tid] = bb;
        const float w = bb.z - bb.x, h = bb.w - bb.y;
        float4 bc; bc.x = (bb.x + bb.z) * 0.5f; bc.y = (bb.y + bb.w) * 0.5f; bc.z = w; bc.w = h;
        sbox_c[tid] = bc;
        sarea[tid]  = w * h;
        slab[tid]   = labels[(size_t)i * M_OBJ + tid];
        bestkey[tid] = 0ull;
    }
    if (tid == 0) s_npos = 0;
    __syncthreads();

    // Phase 1: per-prior IoU vs all objects; argmax over m (first-max) and
    // per-m argmax over p (packed-key atomicMax, first-max via ~p tiebreak).
    for (int p = tid; p < P_PRI; p += TPB) {
        const float4 pc = ((const float4*)priors)[p];
        const float hw = pc.z * 0.5f, hh = pc.w * 0.5f;
        const float px1 = pc.x - hw, py1 = pc.y - hh;
        const float px2 = pc.x + hw, py2 = pc.y + hh;
        const float parea = pc.z * pc.w;
        float best = -1.0f; int bm = 0;
        #pragma unroll
        for (int m = 0; m < M_OBJ; ++m) {
            const float4 b = sbox_xy[m];
            const float lx = fmaxf(b.x, px1), ly = fmaxf(b.y, py1);
            const float hx = fminf(b.z, px2), hy = fminf(b.w, py2);
            const float iw = fmaxf(hx - lx, 0.0f), ih = fmaxf(hy - ly, 0.0f);
            const float inter = iw * ih;
            const float iou = inter / (sarea[m] + parea - inter);
            if (iou > best) { best = iou; bm = m; }
            const unsigned long long key =
                ((unsigned long long)__float_as_uint(iou) << 32) |
                (unsigned long long)(0xFFFFFFFFu - (unsigned)p);
            atomicMax(&bestkey[m], key);
        }
        obj_s[p] = (unsigned char)bm;
        cls_s[p] = (best < 0.5f) ? (unsigned char)0 : (unsigned char)slab[bm];
    }
    __syncthreads();

    // Phase 2: force each object onto its best prior (serial -> "last wins").
    if (tid == 0) {
        for (int m = 0; m < M_OBJ; ++m) {
            const unsigned p = 0xFFFFFFFFu - (unsigned)(bestkey[m] & 0xFFFFFFFFull);
            obj_s[p] = (unsigned char)m;
        }
    }
    __syncthreads();

    // Phase 3: target encode, L1 partials, CE (log-softmax) partials.
    float locpart = 0.0f, confpos = 0.0f;
    int nposl = 0;
    for (int p = tid; p < P_PRI; p += TPB) {
        const float4 pc = ((const float4*)priors)[p];
        const int m = obj_s[p];
        const float4 bc = sbox_c[m];
        const float g0 = 10.0f * (bc.x - pc.x) / pc.z;
        const float g1 = 10.0f * (bc.y - pc.y) / pc.w;
        const float g2 = 5.0f * __logf(bc.z / pc.z);
        const float g3 = 5.0f * __logf(bc.w / pc.w);
        const int cls = cls_s[p];
        const bool pos = (cls != 0);

        const float4 pl = ((const float4*)(locs + ((size_t)i * P_PRI + p) * 4))[0];
        if (pos) {
            locpart += fabsf(pl.x - g0) + fabsf(pl.y - g1) +
                       fabsf(pl.z - g2) + fabsf(pl.w - g3);
            ++nposl;
        }

        const float* sc = scores + ((size_t)i * P_PRI + p) * C_CLS;
        __builtin_prefetch(sc + (size_t)TPB * C_CLS, 0, 1);   // next tile of the stream
        float v[C_CLS];
        float mx = sc[0]; v[0] = mx;
        #pragma unroll
        for (int c = 1; c < C_CLS; ++c) { v[c] = sc[c]; mx = fmaxf(mx, v[c]); }
        float se = 0.0f;
        #pragma unroll
        for (int c = 0; c < C_CLS; ++c) se += __expf(v[c] - mx);
        const float conf = mx + __logf(se) - v[cls];

        if (pos) confpos += conf;
        conf_neg[(size_t)i * P_PRI + p] = pos ? 0.0f : conf;
    }
    atomicAdd(&s_npos, nposl);
    red1[tid] = locpart;
    red2[tid] = confpos;
    __syncthreads();

    if (tid < 32) {      // wave 0 fully active -> EXEC all ones for WMMA
        float s1 = 0.0f, s2 = 0.0f;
        #pragma unroll
        for (int j = 0; j < 8; ++j) { s1 += red1[tid + 32 * j]; s2 += red2[tid + 32 * j]; }
        float t1, t2;
        wave_fold2_wmma(s1, s2, t1, t2);
        if (tid == 0) {
            loc_sum[i]      = t1;
            conf_pos_sum[i] = t2;
            npos_out[i]     = s_npos;
        }
    }
}

// Kernel B: exact top-k sum per image via 8-bit radix select on f32 bits
// (all values >= 0 so uint order == float order). One block per image.
__global__ void __launch_bounds__(TPB)
mbl_hardneg(const float* __restrict__ conf_neg,  // (N,P)
            const int*   __restrict__ npos,      // (N)
            float* __restrict__ hard_sum)        // (N)
{
    const int i   = blockIdx.x;
    const int tid = threadIdx.x;

    __shared__ unsigned hist[256];
    __shared__ unsigned sh_pref, sh_krem;
    __shared__ float red1[TPB];

    int k = 3 * npos[i];
    if (k > P_PRI) k = P_PRI;
    if (k <= 0) { if (tid == 0) hard_sum[i] = 0.0f; return; }

    unsigned pref = 0u, krem = (unsigned)k;
    for (int shift = 24; shift >= 0; shift -= 8) {
        hist[tid] = 0u;
        __syncthreads();
        const unsigned hmask = (shift == 24) ? 0u : (0xFFFFFFFFu << (shift + 8));
        for (int p = tid; p < P_PRI; p += TPB) {
            const unsigned v = __float_as_uint(conf_neg[(size_t)i * P_PRI + p]);
            if ((v & hmask) == pref) atomicAdd(&hist[(v >> shift) & 255u], 1u);
        }
        __syncthreads();
        if (tid == 0) {
            unsigned cum = 0u;
            for (int b = 255; b >= 0; --b) {
                const unsigned h = hist[b];
                if (cum + h >= krem) {
                    sh_pref = pref | ((unsigned)b << shift);
                    sh_krem = krem - cum;
                    break;
                }
                cum += h;
            }
        }
        __syncthreads();
        pref = sh_pref; krem = sh_krem;
        __syncthreads();
    }

    // pref == bits of the k-th largest value T; krem copies of T are in the top-k.
    const float T = __uint_as_float(pref);
    float s = 0.0f;
    for (int p = tid; p < P_PRI; p += TPB) {
        const float v = conf_neg[(size_t)i * P_PRI + p];
        if (v > T) s += v;
    }
    red1[tid] = s;
    __syncthreads();
    if (tid < 32) {
        float s1 = 0.0f;
        #pragma unroll
        for (int j = 0; j < 8; ++j) s1 += red1[tid + 32 * j];
        float t1, t2;
        wave_fold2_wmma(s1, 0.0f, t1, t2);
        if (tid == 0) hard_sum[i] = t1 + (float)krem * T;
    }
}

// Kernel C: final scalar combine (serial, deterministic).
__global__ void mbl_finalize(const float* __restrict__ loc_sum,
                             const float* __restrict__ conf_pos_sum,
                             const float* __restrict__ hard_sum,
                             const int*   __restrict__ npos,
                             float* __restrict__ out)
{
    if (threadIdx.x == 0 && blockIdx.x == 0) {
        float np = 0.0f, conf = 0.0f, loc = 0.0f;
        for (int i = 0; i < N_IMG; ++i) {
            np   += (float)npos[i];
            conf += conf_pos_sum[i] + hard_sum[i];
            loc  += loc_sum[i];
        }
        out[0] = conf / np + loc / (np * 4.0f);
    }
}

extern "C" void kernel_launch(void* const* d_in, const int* in_sizes, int n_in,
                              void* d_out, int out_size, void* d_ws, size_t ws_size,
                              hipStream_t stream) {
    const float* locs   = (const float*)d_in[0];   // (N,P,4)
    const float* scores = (const float*)d_in[1];   // (N,P,C)
    const float* boxes  = (const float*)d_in[2];   // (N,M,4)
    const int*   labels = (const int*)d_in[3];     // (N,M)
    const float* priors = (const float*)d_in[4];   // (P,4)

    float* conf_neg = (float*)d_ws;                          // N*P
    float* loc_sum  = conf_neg + (size_t)N_IMG * P_PRI;      // N
    float* conf_pos = loc_sum + N_IMG;                       // N
    float* hard_sum = conf_pos + N_IMG;                      // N
    int*   npos     = (int*)(hard_sum + N_IMG);              // N

    mbl_assign_conf<<<N_IMG, TPB, 0, stream>>>(locs, scores, boxes, labels, priors,
                                               conf_neg, loc_sum, conf_pos, npos);
    mbl_hardneg<<<N_IMG, TPB, 0, stream>>>(conf_neg, npos, hard_sum);
    mbl_finalize<<<1, 32, 0, stream>>>(loc_sum, conf_pos, hard_sum, npos,
                                       (float*)d_out);
}